// DetectionOut_43885975830749
// MI455X (gfx1250) — compile-verified
//
#include <hip/hip_runtime.h>
#include <hip/hip_bf16.h>
#include <stdint.h>

#define NBATCH 32
#define NPRI   65536
#define NCH    84
#define NROWS  256          // priors per block in score pass
#define TOPK   400
#define KEEPK  200
#define CAP    2048         // candidate buffer for top-k
#define KW     13           // ceil(400/32) bitmask words
#define CONF_THRESH 0.5f
#define NMS_THRESH  0.5f
#define VAR_C  0.1f
#define VAR_S  0.2f

typedef float v2f __attribute__((ext_vector_type(2)));
typedef float v8f __attribute__((ext_vector_type(8)));
typedef int   vi4 __attribute__((vector_size(16)));
typedef __attribute__((address_space(1))) vi4 as1_vi4;
typedef __attribute__((address_space(3))) vi4 as3_vi4;

// ---- CDNA5 feature detection (device pass only; host pass takes fallbacks) ----
#if defined(__HIP_DEVICE_COMPILE__) && defined(__has_builtin)
#  if __has_builtin(__builtin_amdgcn_global_load_async_to_lds_b128)
#    define USE_ASYNC_LDS 1
#  endif
#  if __has_builtin(__builtin_amdgcn_wmma_f32_16x16x4_f32)
#    define USE_WMMA 1
#  endif
#endif
#ifndef USE_ASYNC_LDS
#define USE_ASYNC_LDS 0
#endif
#ifndef USE_WMMA
#define USE_WMMA 0
#endif

__device__ __forceinline__ uint32_t f2sort(float f) {
  uint32_t s = __float_as_uint(f);
  return (s & 0x80000000u) ? ~s : (s | 0x80000000u);
}

// =====================================================================
// K1: stream predictions through LDS (async-to-LDS), per-prior masked
//     max-score -> sortable u32 key.  grid (NPRI/NROWS, NBATCH), 256 thr
// =====================================================================
__global__ void k_score(const float* __restrict__ pred,
                        uint32_t* __restrict__ keys) {
  const int b   = blockIdx.y;
  const int p0  = blockIdx.x * NROWS;
  const int tid = threadIdx.x;
  extern __shared__ float tile[];            // NROWS * NCH floats (86016 B, 16B aligned rows)
  const float* src = pred + ((size_t)b * NPRI + (size_t)p0) * NCH;

#if USE_ASYNC_LDS
  // tile region is fully contiguous in global memory: NROWS*NCH*4 = 86016 B = 5376 x 16B
  const int chunks = NROWS * NCH / 4;        // b128 chunks
  for (int k = tid; k < chunks; k += NROWS) {
    const char* g = (const char*)src + (size_t)k * 16;
    char*       l = (char*)tile + (size_t)k * 16;
    __builtin_amdgcn_global_load_async_to_lds_b128(
        (as1_vi4*)(const_cast<char*>(g)),
        (as3_vi4*)(l), 0, 0);
  }
#  if __has_builtin(__builtin_amdgcn_s_wait_asynccnt)
  __builtin_amdgcn_s_wait_asynccnt(0);
#  else
  asm volatile("s_wait_asynccnt 0" ::: "memory");
#  endif
#else
  const int vecs = NROWS * NCH / 4;          // float4 chunks
  for (int k = tid; k < vecs; k += NROWS)
    ((float4*)tile)[k] = ((const float4*)src)[k];
#endif
  __syncthreads();

  const float* row = tile + tid * NCH;
  float m = row[4];
  #pragma unroll 4
  for (int c = 5; c < NCH; ++c) m = fmaxf(m, row[c]);
  float masked = (m > CONF_THRESH) ? m : -__builtin_inff();
  keys[(size_t)b * NPRI + p0 + tid] = f2sort(masked);
}

// =====================================================================
// K2: per-batch top-400 via 4-level radix select + bitonic sort of ties.
//     grid NBATCH, 1024 threads
// =====================================================================
__global__ void k_topk(const uint32_t* __restrict__ keys,
                       uint32_t* __restrict__ sel) {
  const int b   = blockIdx.x;
  const int tid = threadIdx.x;
  const uint32_t* kb = keys + (size_t)b * NPRI;

  __shared__ uint32_t hist[256];
  __shared__ uint32_t sPrefix, sRemain, sCount;
  __shared__ unsigned long long cand[CAP];

  if (tid == 0) { sPrefix = 0u; sRemain = TOPK; }
  __syncthreads();

  for (int level = 0; level < 4; ++level) {
    const int shift = 24 - 8 * level;
    for (int i = tid; i < 256; i += blockDim.x) hist[i] = 0u;
    __syncthreads();
    const uint32_t pref = sPrefix;
    for (int i = tid; i < NPRI; i += blockDim.x) {
      uint32_t u = kb[i];
      if (((uint64_t)u >> (shift + 8)) == ((uint64_t)pref >> (shift + 8)))
        atomicAdd(&hist[(u >> shift) & 255u], 1u);
    }
    __syncthreads();
    if (tid == 0) {
      uint32_t rem = sRemain;
      int bsel = 0;
      for (int bin = 255; bin >= 0; --bin) {
        uint32_t c = hist[bin];
        if (c >= rem) { bsel = bin; break; }
        rem -= c;
      }
      sPrefix = pref | ((uint32_t)bsel << shift);
      sRemain = rem;
    }
    __syncthreads();
  }

  const uint32_t T = sPrefix;     // exact key value of rank-400 element
  if (tid == 0) sCount = 0u;
  __syncthreads();
  for (int i = tid; i < NPRI; i += blockDim.x) {
    uint32_t u = kb[i];
    if (u >= T) {
      uint32_t pos = atomicAdd(&sCount, 1u);
      if (pos < CAP)
        cand[pos] = ((unsigned long long)u << 32) | (uint32_t)(~(uint32_t)i);
    }
  }
  __syncthreads();
  uint32_t cnt = sCount; if (cnt > CAP) cnt = CAP;
  for (int i = tid; i < CAP; i += blockDim.x)
    if (i >= (int)cnt) cand[i] = 0ull;
  __syncthreads();

  // bitonic sort descending by (key, ~idx)  -> stable top_k order
  for (int k = 2; k <= CAP; k <<= 1) {
    for (int j = k >> 1; j > 0; j >>= 1) {
      for (int i = tid; i < CAP; i += blockDim.x) {
        int ixj = i ^ j;
        if (ixj > i) {
          bool desc = ((i & k) == 0);
          unsigned long long a = cand[i], c = cand[ixj];
          if (desc ? (a < c) : (a > c)) { cand[i] = c; cand[ixj] = a; }
        }
      }
      __syncthreads();
    }
  }
  for (int r = tid; r < TOPK; r += blockDim.x)
    sel[b * TOPK + r] = ~(uint32_t)(cand[r] & 0xFFFFFFFFull);
}

// =====================================================================
// K3: gather selected rows, decode boxes, per-row label/score/valid.
//     grid NBATCH, 128 threads
// =====================================================================
__global__ void k_gather(const float* __restrict__ pred,
                         const float* __restrict__ priors,
                         const uint32_t* __restrict__ sel,
                         float* __restrict__ cbox, int* __restrict__ clabel,
                         float* __restrict__ cscore, uint32_t* __restrict__ cvalid) {
  const int b = blockIdx.x;
  for (int r = threadIdx.x; r < TOPK; r += blockDim.x) {
    uint32_t p = sel[b * TOPK + r];
    const float* row = pred + ((size_t)b * NPRI + p) * NCH;
    float l0 = row[0], l1 = row[1], l2 = row[2], l3 = row[3];
    float px = priors[p * 4 + 0], py = priors[p * 4 + 1];
    float pw = priors[p * 4 + 2], ph = priors[p * 4 + 3];
    float cx = px + l0 * VAR_C * pw;
    float cy = py + l1 * VAR_C * ph;
    float w  = pw * expf(l2 * VAR_S);
    float h  = ph * expf(l3 * VAR_S);
    float m = row[4]; int am = 0;
    for (int c = 5; c < NCH; ++c) { float v = row[c]; if (v > m) { m = v; am = c - 4; } }
    int o = b * TOPK + r;
    cbox[o * 4 + 0] = cx - 0.5f * w;
    cbox[o * 4 + 1] = cy - 0.5f * h;
    cbox[o * 4 + 2] = cx + 0.5f * w;
    cbox[o * 4 + 3] = cy + 0.5f * h;
    clabel[o] = am;
    cscore[o] = m;
    cvalid[o] = (m > CONF_THRESH) ? 1u : 0u;
  }
}

// =====================================================================
// K4: NMS (400x400 suppression bitmatrix, WMMA pairwise area sums),
//     keep-scan, argsort by ymin, emit 200x6.  grid NBATCH, 512 threads
// =====================================================================
__global__ void k_nms(const float* __restrict__ cbox, const int* __restrict__ clabel,
                      const float* __restrict__ cscore, const uint32_t* __restrict__ cvalid,
                      float* __restrict__ out) {
  const int b   = blockIdx.x;
  const int tid = threadIdx.x;
  __shared__ float    bx[TOPK][4];
  __shared__ float    areaS[TOPK];
  __shared__ float    scoreS[TOPK];
  __shared__ int      labelS[TOPK];
  __shared__ uint32_t sup[TOPK][KW];
  __shared__ uint32_t keep[KW];
  __shared__ unsigned long long skey[512];

  for (int r = tid; r < TOPK; r += blockDim.x) {
    int o = b * TOPK + r;
    float x0 = cbox[o * 4 + 0], y0 = cbox[o * 4 + 1];
    float x1 = cbox[o * 4 + 2], y1 = cbox[o * 4 + 3];
    bx[r][0] = x0; bx[r][1] = y0; bx[r][2] = x1; bx[r][3] = y1;
    areaS[r] = (x1 - x0) * (y1 - y0);
    scoreS[r] = cscore[o];
    labelS[r] = clabel[o];
  }
  for (int i = tid; i < TOPK * KW; i += blockDim.x) ((uint32_t*)sup)[i] = 0u;
  if (tid < KW) keep[tid] = 0u;
  __syncthreads();
  for (int r = tid; r < TOPK; r += blockDim.x)
    if (cvalid[b * TOPK + r]) atomicOr(&keep[r >> 5], 1u << (r & 31));
  __syncthreads();

  // ---- suppression matrix in 16x16 tiles, one wave per tile ----
  const int lane = tid & 31;
  const int wid  = tid >> 5;
  const int nw   = blockDim.x >> 5;
  for (int t = wid; t < 25 * 25; t += nw) {
    const int ti = t / 25, tj = t % 25;
    const int i0 = ti * 16, j0 = tj * 16;
    v8f d;
#if USE_WMMA
    // D = area_col x e0_row  (gives area_i), then += e0_col x area_row (gives +area_j)
    v2f a, bb, a2, b2;
    a.x  = (lane < 16) ? areaS[i0 + lane] : 0.f;  a.y  = 0.f;
    bb.x = (lane < 16) ? 1.f : 0.f;               bb.y = 0.f;
    v8f c0 = {0.f, 0.f, 0.f, 0.f, 0.f, 0.f, 0.f, 0.f};
    d = __builtin_amdgcn_wmma_f32_16x16x4_f32(false, a, false, bb,
                                              (short)0, c0, false, false);
    a2.x = (lane < 16) ? 1.f : 0.f;               a2.y = 0.f;
    b2.x = (lane < 16) ? areaS[j0 + lane] : 0.f;  b2.y = 0.f;
    d = __builtin_amdgcn_wmma_f32_16x16x4_f32(false, a2, false, b2,
                                              (short)0, d, false, false);
#else
    for (int v = 0; v < 8; ++v) {
      int i = i0 + v + ((lane >= 16) ? 8 : 0);
      int j = j0 + (lane & 15);
      d[v] = areaS[i] + areaS[j];
    }
#endif
    const int j = j0 + (lane & 15);
    const float jx0 = bx[j][0], jy0 = bx[j][1], jx1 = bx[j][2], jy1 = bx[j][3];
    #pragma unroll
    for (int v = 0; v < 8; ++v) {
      int i = i0 + v + ((lane >= 16) ? 8 : 0);
      float ltx = fmaxf(bx[i][0], jx0);
      float lty = fmaxf(bx[i][1], jy0);
      float rbx = fminf(bx[i][2], jx1);
      float rby = fminf(bx[i][3], jy1);
      float iw = fmaxf(rbx - ltx, 0.f);
      float ih = fmaxf(rby - lty, 0.f);
      float inter = iw * ih;
      float iou = inter / (d[v] - inter + 1e-9f);
      uint32_t m = (uint32_t)__ballot(iou > NMS_THRESH);
      if (lane == 0) {
        int w  = j0 >> 5;
        int sh = j0 & 31;                         // 0 or 16
        atomicOr(&sup[i0 + v][w],     (m & 0xFFFFu) << sh);
        atomicOr(&sup[i0 + v + 8][w], (m >> 16)     << sh);
      }
    }
  }
  __syncthreads();

  // ---- sequential keep propagation (matches fori_loop semantics) ----
  if (tid == 0) {
    uint32_t kp[KW];
    for (int w = 0; w < KW; ++w) kp[w] = keep[w];
    for (int i = 0; i < TOPK; ++i) {
      if ((kp[i >> 5] >> (i & 31)) & 1u) {
        const int wi = i >> 5, bi = i & 31;
        for (int w = wi; w < KW; ++w) {
          uint32_t s = sup[i][w];
          if (w == wi) s &= (bi == 31) ? 0u : (0xFFFFFFFFu << (bi + 1)); // only j > i
          kp[w] &= ~s;
        }
      }
    }
    for (int w = 0; w < KW; ++w) keep[w] = kp[w];
  }
  __syncthreads();

  // ---- stable argsort ascending by (kept ? ymin : +inf, idx) ----
  {
    unsigned long long key;
    if (tid < TOPK) {
      bool kpt = (keep[tid >> 5] >> (tid & 31)) & 1u;
      float kf = kpt ? bx[tid][1] : __builtin_inff();
      key = ((unsigned long long)f2sort(kf) << 32) | (uint32_t)tid;
    } else {
      key = ((unsigned long long)f2sort(__builtin_inff()) << 32) | (uint32_t)tid;
    }
    skey[tid] = key;
  }
  __syncthreads();
  for (int k = 2; k <= 512; k <<= 1) {
    for (int j = k >> 1; j > 0; j >>= 1) {
      int i = tid, ixj = i ^ j;
      if (ixj > i) {
        bool asc = ((i & k) == 0);
        unsigned long long aa = skey[i], cc = skey[ixj];
        if (asc ? (aa > cc) : (aa < cc)) { skey[i] = cc; skey[ixj] = aa; }
      }
      __syncthreads();
    }
  }
  if (tid < KEEPK) {
    uint32_t o = (uint32_t)(skey[tid] & 0xFFFFFFFFull);
    bool kpt = (o < TOPK) && (((keep[o >> 5] >> (o & 31)) & 1u) != 0u);
    float* dst = out + ((size_t)b * KEEPK + tid) * 6;
    if (kpt) {
      dst[0] = bx[o][0]; dst[1] = bx[o][1]; dst[2] = bx[o][2]; dst[3] = bx[o][3];
      dst[4] = (float)labelS[o]; dst[5] = scoreS[o];
    } else {
      dst[0] = 0.f; dst[1] = 0.f; dst[2] = 0.f;
      dst[3] = 0.f; dst[4] = 0.f; dst[5] = 0.f;
    }
  }
}

// =====================================================================
extern "C" void kernel_launch(void* const* d_in, const int* in_sizes, int n_in,
                              void* d_out, int out_size, void* d_ws, size_t ws_size,
                              hipStream_t stream) {
  const float* pred   = (const float*)d_in[0];   // 32 x 65536 x 84
  const float* priors = (const float*)d_in[1];   // 65536 x 4
  float* out = (float*)d_out;                    // 32 x 200 x 6
  char* ws = (char*)d_ws;

  // workspace layout (bytes)
  uint32_t* keys = (uint32_t*)(ws);                        // 32*65536*4 = 8388608
  uint32_t* sel  = (uint32_t*)(ws + 8388608);              // 32*400*4   = 51200
  float*    cbox = (float*)   (ws + 8439808);              // 32*400*16  = 204800
  int*      clab = (int*)     (ws + 8644608);              // 51200
  float*    cscr = (float*)   (ws + 8695808);              // 51200
  uint32_t* cval = (uint32_t*)(ws + 8747008);              // 51200

  dim3 g1(NPRI / NROWS, NBATCH);
  k_score<<<g1, NROWS, NROWS * NCH * sizeof(float), stream>>>(pred, keys);
  k_topk<<<NBATCH, 1024, 0, stream>>>(keys, sel);
  k_gather<<<NBATCH, 128, 0, stream>>>(pred, priors, sel, cbox, clab, cscr, cval);
  k_nms<<<NBATCH, 512, 0, stream>>>(cbox, clab, cscr, cval, out);
}